// myRNNModel_38792144618212
// MI455X (gfx1250) — compile-verified
//
#include <hip/hip_runtime.h>

typedef __attribute__((ext_vector_type(2))) float v2f;
typedef __attribute__((ext_vector_type(8))) float v8f;

#define BB 1024
#define TT 256
#define HH 64
#define EE 32
#define VV 10
#define OO 10

#if __has_builtin(__builtin_amdgcn_tanhf)
#define DEV_TANH(x) __builtin_amdgcn_tanhf(x)   // V_TANH_F32 hardware transcendental
#else
#define DEV_TANH(x) tanhf(x)
#endif

// Workspace layout (floats):
//   [0    , 640 )  lut0[10][64] : emb_table @ Wx[:32,:] + b_rnn
//   [640  , 1280)  lut1[10][64] : emb_table @ Wx[32:,:]
//   [1280 , 2304)  wdp[64][16]  : Wd zero-padded 10 -> 16 cols
//   [2304 , 2320)  bdp[16]      : bd zero-padded

__global__ __launch_bounds__(64) void rnn_prep_kernel(
    const float* __restrict__ emb, const float* __restrict__ Wx,
    const float* __restrict__ brnn, const float* __restrict__ Wd,
    const float* __restrict__ bd, float* __restrict__ ws) {
  const int j = threadIdx.x;  // 0..63 : one hidden column each
  for (int v = 0; v < VV; ++v) {
    float s0 = brnn[j], s1 = 0.f;
    for (int d = 0; d < EE; ++d) {
      const float e = emb[v * EE + d];
      s0 += e * Wx[d * HH + j];
      s1 += e * Wx[(EE + d) * HH + j];
    }
    ws[v * HH + j] = s0;
    ws[VV * HH + v * HH + j] = s1;
  }
  for (int n = 0; n < 16; ++n)
    ws[1280 + j * 16 + n] = (n < OO) ? Wd[j * OO + n] : 0.f;
  if (j < 16) ws[2304 + j] = (j < OO) ? bd[j] : 0.f;
}

// One wave (32 threads) per block; each wave owns 16 batch rows and runs the
// whole T=256 scan with Wh/Wd resident in VGPRs as WMMA B-fragments.
__global__ __launch_bounds__(32) void rnn_scan_kernel(
    const int* __restrict__ num1, const int* __restrict__ num2,
    const float* __restrict__ Wh, const float* __restrict__ ws,
    float* __restrict__ out) {
  __shared__ float lut[2 * VV * HH];  // 1280 floats
  __shared__ float hbuf[16 * HH];     // h state, 16 rows x 64, row-major
  __shared__ int   vidx[2][16];       // this step's digits for the 16 rows

  const int lane = threadIdx.x;
  const int half = lane >> 4;   // 0: lanes 0-15, 1: lanes 16-31
  const int m    = lane & 15;
  const int kb   = half * 2;    // K sub-row base inside a 4-wide K step
  const int row_base = blockIdx.x * 16;

  for (int i = lane; i < 2 * VV * HH; i += 32) lut[i] = ws[i];
  for (int i = lane; i < 16 * HH; i += 32) hbuf[i] = 0.f;

  // Wh as WMMA B-fragments (4x16 per fragment): b.x = row(4k+kb), b.y = row(4k+kb+1), col n*16+m
  v2f whb[16][4];
#pragma unroll
  for (int k = 0; k < 16; ++k)
#pragma unroll
    for (int n = 0; n < 4; ++n) {
      whb[k][n].x = Wh[(4 * k + kb + 0) * HH + n * 16 + m];
      whb[k][n].y = Wh[(4 * k + kb + 1) * HH + n * 16 + m];
    }
  // Padded Wd B-fragments (single N-tile)
  v2f wdb[16];
#pragma unroll
  for (int k = 0; k < 16; ++k) {
    wdb[k].x = ws[1280 + (4 * k + kb + 0) * 16 + m];
    wdb[k].y = ws[1280 + (4 * k + kb + 1) * 16 + m];
  }
  const float bdv = ws[2304 + m];

  // A-fragments of h (h starts at zero)
  v2f a[16];
#pragma unroll
  for (int k = 0; k < 16; ++k) { a[k].x = 0.f; a[k].y = 0.f; }

  __syncthreads();

  const int* __restrict__ nsrc = half ? num2 : num1;

  for (int t = 0; t < TT; ++t) {
    // lanes 0-15 fetch num1 for their row, lanes 16-31 fetch num2
    vidx[half][m] = nsrc[(row_base + m) * TT + t];
    if (t + 1 < TT) __builtin_prefetch(&nsrc[(row_base + m) * TT + t + 1], 0, 3);
    __syncthreads();

    // C init = xproj(t) gathered from the LUTs (D layout: VGPR r -> row r+8*half, col n*16+m)
    v8f acc[4];
#pragma unroll
    for (int r = 0; r < 8; ++r) {
      const int hr = r + 8 * half;
      const int d1 = vidx[0][hr];
      const int d2 = vidx[1][hr];
#pragma unroll
      for (int n = 0; n < 4; ++n)
        acc[n][r] = lut[d1 * HH + n * 16 + m] + lut[VV * HH + d2 * HH + n * 16 + m];
    }

    // h_old @ Wh : M=16, N=64, K=64 -> 64x V_WMMA_F32_16X16X4_F32
#pragma unroll
    for (int k = 0; k < 16; ++k)
#pragma unroll
      for (int n = 0; n < 4; ++n)
        acc[n] = __builtin_amdgcn_wmma_f32_16x16x4_f32(
            false, a[k], false, whb[k][n], (short)0, acc[n], false, false);

    // tanh (hardware V_TANH_F32), write h_new back to LDS (D layout -> row-major)
#pragma unroll
    for (int n = 0; n < 4; ++n)
#pragma unroll
      for (int r = 0; r < 8; ++r)
        hbuf[(r + 8 * half) * HH + n * 16 + m] = DEV_TANH(acc[n][r]);
    __syncthreads();

    // Reload A-fragments of h_new: feeds the logits GEMM AND the next timestep
#pragma unroll
    for (int k = 0; k < 16; ++k)
      a[k] = *(const v2f*)&hbuf[m * HH + 4 * k + kb];

    // logits tile: h_new(16x64) @ WdPad(64x16) + bd
    v8f c2;
#pragma unroll
    for (int r = 0; r < 8; ++r) c2[r] = bdv;
#pragma unroll
    for (int k = 0; k < 16; ++k)
      c2 = __builtin_amdgcn_wmma_f32_16x16x4_f32(
          false, a[k], false, wdb[k], (short)0, c2, false, false);

    if (m < OO) {
#pragma unroll
      for (int r = 0; r < 8; ++r) {
        const int row = row_base + r + 8 * half;
        out[(size_t)row * TT * OO + (size_t)t * OO + m] = c2[r];
      }
    }
    // no trailing barrier needed: LDS ops from a single wave complete in order
  }
}

extern "C" void kernel_launch(void* const* d_in, const int* in_sizes, int n_in,
                              void* d_out, int out_size, void* d_ws, size_t ws_size,
                              hipStream_t stream) {
  const int*   num1 = (const int*)d_in[0];
  const int*   num2 = (const int*)d_in[1];
  const float* emb  = (const float*)d_in[2];
  const float* Wx   = (const float*)d_in[3];
  const float* Wh   = (const float*)d_in[4];
  const float* brnn = (const float*)d_in[5];
  const float* Wd   = (const float*)d_in[6];
  const float* bd   = (const float*)d_in[7];
  float* out = (float*)d_out;
  float* ws  = (float*)d_ws;

  rnn_prep_kernel<<<1, 64, 0, stream>>>(emb, Wx, brnn, Wd, bd, ws);
  rnn_scan_kernel<<<BB / 16, 32, 0, stream>>>(num1, num2, Wh, ws, out);
}